// SUBNCEL_46205258170686
// MI455X (gfx1250) — compile-verified
//
#include <hip/hip_runtime.h>

typedef __attribute__((ext_vector_type(2))) float    v2f;
typedef __attribute__((ext_vector_type(8))) float    v8f;
typedef __attribute__((ext_vector_type(4))) int      v4i;
typedef __attribute__((ext_vector_type(4))) unsigned u32x4;
typedef __attribute__((ext_vector_type(8))) int      i32x8;
typedef __attribute__((ext_vector_type(4))) int      i32x4;

typedef __attribute__((address_space(1))) v4i gv4i;  // global int4
typedef __attribute__((address_space(3))) v4i lv4i;  // LDS int4

#define BB 2048
#define CC 16
#define DD 256
#define OUTW 353            // D + 6*C + 1
#define THRED 0.8f
#define EPSV 1e-8f

#if __has_builtin(__builtin_amdgcn_tensor_load_to_lds)
#define HAVE_TDM 1
#endif
#if __has_builtin(__builtin_amdgcn_global_load_async_to_lds_b128)
#define ASYNC_BUILTIN 1
#endif

// Async global -> LDS, 16 bytes per lane, no VGPR staging (ASYNCcnt-tracked).
__device__ __forceinline__ void async_g2l_b128(const float* __restrict__ g,
                                               float* l) {
#ifdef ASYNC_BUILTIN
  __builtin_amdgcn_global_load_async_to_lds_b128(
      (gv4i*)g, (lv4i*)l, /*offset=*/0, /*cpol=*/0);
#else
  unsigned lds_off =
      (unsigned)(size_t)(__attribute__((address_space(3))) void*)l;
  asm volatile("global_load_async_to_lds_b128 %0, %1, off"
               :
               : "v"(lds_off), "v"(g)
               : "memory");
#endif
}

__device__ __forceinline__ void wait_async0() {
#if __has_builtin(__builtin_amdgcn_s_wait_asynccnt)
  __builtin_amdgcn_s_wait_asynccnt(0);
#else
  asm volatile("s_wait_asynccnt 0" ::: "memory");
#endif
}

__device__ __forceinline__ void wait_tensor0() {
#if __has_builtin(__builtin_amdgcn_s_wait_tensorcnt)
  __builtin_amdgcn_s_wait_tensorcnt(0);
#else
  asm volatile("s_wait_tensorcnt 0" ::: "memory");
#endif
}

#ifdef HAVE_TDM
// One TDM descriptor: copy 16x256 f32 tile (contiguous rows of 1KB) from
// global VA `ga` into LDS at byte offset `lds_off`, padding +4 DWORDs after
// every 256 DWORDs so the LDS row stride becomes 260 floats.
__device__ __forceinline__ void tdm_tile_load(unsigned long long ga,
                                              unsigned lds_off) {
  u32x4 g0;
  g0.x = 1u;                                            // count=1, user mode
  g0.y = lds_off;                                       // lds_addr [63:32]
  g0.z = (unsigned)ga;                                  // global_addr[31:0]
  g0.w = (unsigned)((ga >> 32) & 0x1FFFFFFu) | (2u << 30);  // addr[56:32], type=2
  i32x8 g1;
  g1[0] = (int)((2u << 16)      // data_size = 4B
              | (1u << 20)      // pad_enable
              | (7u << 22)      // pad_interval: 256 DWORDs
              | (3u << 25));    // pad_amount: 4 DWORDs
  g1[1] = (int)(256u << 16);    // tensor_dim0[15:0]=256 in bits [63:48]
  g1[2] = (int)(16u << 16);     // tensor_dim0[31:16]=0, tensor_dim1[15:0]=16
  g1[3] = (int)(256u << 16);    // tensor_dim1[31:16]=0, tile_dim0=256
  g1[4] = 16;                   // tile_dim1=16, tile_dim2=0
  g1[5] = 256;                  // tensor_dim0_stride[31:0]=256
  g1[6] = 0;                    // stride0 hi, stride1 lo
  g1[7] = 0;                    // stride1 hi
  i32x4 gz = {0, 0, 0, 0};      // groups 2/3 unused (tile_dim3/4 = 0)
#if __clang_major__ >= 23
  i32x8 gz8 = {0, 0, 0, 0, 0, 0, 0, 0};
  __builtin_amdgcn_tensor_load_to_lds(g0, g1, gz, gz, gz8, 0);
#else
  __builtin_amdgcn_tensor_load_to_lds(g0, g1, gz, gz, 0);
#endif
}
#endif

// Shared layout (float units), rows padded to 260 for bank-conflict-free b64 reads:
//   sA     [16][260]      @ 0       (center tile)
//   sB     [6][16][260]   @ 4160    (neighbor tiles, parts: -1,-2,-3,+1,+2,+3)
//   sNormC [16]           @ 29120
//   sNormN [6*16]         @ 29136
//   sMask  [6*16]         @ 29232
//   sAdj   [6][16][16]    @ 29328
//   sSum   [16]           @ 30864
// total 30880 floats = 123520 bytes

__global__ __launch_bounds__(256) void graph_wmma_kernel(
    const float* __restrict__ emb, const int* __restrict__ nm,
    float* __restrict__ out) {
  extern __shared__ float smem[];
  float* sA     = smem;
  float* sB     = smem + 4160;
  float* sNormC = smem + 29120;
  float* sNormN = smem + 29136;
  float* sMask  = smem + 29232;
  float* sAdj   = smem + 29328;
  float* sSum   = smem + 30864;

  const int b   = blockIdx.x;
  const int tid = threadIdx.x;

  // ---- Phase 1: stage 7 tiles (center + 6 shifted) into LDS ----
  // tile t: 0 = center; 1..3 -> offsets -1,-2,-3; 4..6 -> +1,+2,+3
#ifdef HAVE_TDM
  if (tid < 32) {  // wave 0 issues the TDM descriptors (EXEC-independent ops)
    for (int t = 0; t < 7; ++t) {
      int off = (t == 0) ? 0 : ((t <= 3) ? -t : (t - 3));
      int gb  = b + off;
      if (gb >= 0 && gb < BB) {
        float* ldst = (t == 0) ? sA : (sB + (t - 1) * 4160);
        unsigned lds_off =
            (unsigned)(size_t)(__attribute__((address_space(3))) void*)ldst;
        unsigned long long ga =
            (unsigned long long)(const void*)(emb + (size_t)gb * (CC * DD));
        tdm_tile_load(ga, lds_off);
      }
    }
    wait_tensor0();
  }
  // cooperative zero-fill of out-of-range boundary tiles only
  for (int idx = tid; idx < 7 * 16 * 64; idx += 256) {
    int t   = idx >> 10;
    int off = (t == 0) ? 0 : ((t <= 3) ? -t : (t - 3));
    int gb  = b + off;
    if (gb < 0 || gb >= BB) {
      int rem = idx & 1023;
      int row = rem >> 6;
      int c4  = rem & 63;
      float* dst = ((t == 0) ? sA : (sB + (t - 1) * 4160)) + row * 260 + c4 * 4;
      *(float4*)dst = make_float4(0.f, 0.f, 0.f, 0.f);
    }
  }
#else
  for (int idx = tid; idx < 7 * 16 * 64; idx += 256) {
    int t   = idx >> 10;
    int rem = idx & 1023;
    int row = rem >> 6;
    int c4  = rem & 63;
    int off = (t == 0) ? 0 : ((t <= 3) ? -t : (t - 3));
    int gb  = b + off;
    float* dst = ((t == 0) ? sA : (sB + (t - 1) * 4160)) + row * 260 + c4 * 4;
    if (gb >= 0 && gb < BB) {
      async_g2l_b128(emb + ((size_t)(gb * CC + row)) * DD + c4 * 4, dst);
    } else {
      *(float4*)dst = make_float4(0.f, 0.f, 0.f, 0.f);
    }
  }
#endif

  // ---- Phase 2: cumulative 0/1 masks per (part p, candidate c) ----
  if (tid < 96) {
    int p = tid >> 4, c = tid & 15;
    float m = 0.f;
    if (p < 3) {
      int k = p;
      if (b - k - 1 >= 0) {
        m = 1.f;
        for (int j = 0; j <= k; ++j) m *= (float)nm[(b - 1 - j) * CC + c];
      }
    } else {
      int k = p - 3;
      if (b + k + 1 < BB) {
        m = 1.f;
        for (int j = 0; j <= k; ++j) m *= (float)nm[(b + j) * CC + c];
      }
    }
    sMask[tid] = m;
  }

#ifndef HAVE_TDM
  wait_async0();      // this wave's async tile traffic is in LDS
#endif
  __syncthreads();    // all staging (TDM waited by wave 0) visible to all

  // ---- Phase 3: row L2 norms of all 112 staged rows ----
  if (tid < 112) {
    const float* rowp = (tid < 16) ? (sA + tid * 260) : (sB + (tid - 16) * 260);
    float s = 0.f;
#pragma unroll
    for (int d = 0; d < 256; d += 4) {
      float4 v = *(const float4*)(rowp + d);
      s += v.x * v.x + v.y * v.y + v.z * v.z + v.w * v.w;
    }
    float n = sqrtf(s);
    if (tid < 16) sNormC[tid] = n;
    else          sNormN[tid - 16] = n;
  }
  __syncthreads();

  const int lane = tid & 31;
  const int w    = tid >> 5;   // wave id (wave32), uniform per wave

  // ---- Phase 4: waves 0..5 -> one 16x16 (K=256) GEMM each via WMMA f32 ----
  if (w < 6) {
    const int row  = lane & 15;     // M for A-frag, N for B-frag, N for D
    const int half = lane >> 4;     // selects K pair (A/B) and M high-half (D)
    const float* Ap = sA + row * 260 + 2 * half;
    const float* Bp = sB + w * 4160 + row * 260 + 2 * half;
    v8f acc = {};
#pragma unroll 8
    for (int kk = 0; kk < 64; ++kk) {
      v2f a  = *(const v2f*)(Ap + 4 * kk);
      v2f bv = *(const v2f*)(Bp + 4 * kk);
      acc = __builtin_amdgcn_wmma_f32_16x16x4_f32(
          /*neg_a=*/false, a, /*neg_b=*/false, bv,
          /*c_mod=*/(short)0, acc, /*reuse_a=*/false, /*reuse_b=*/false);
    }
    // D layout: lane<16 -> D[r][lane]; lane>=16 -> D[r+8][lane-16]
    const int   j  = row;
    const float nj = sNormN[w * 16 + j];
    const float mj = sMask[w * 16 + j];
#pragma unroll
    for (int r = 0; r < 8; ++r) {
      int   i   = r + 8 * half;
      float sim = acc[r] / fmaxf(sNormC[i] * nj, EPSV);
      float val = (mj != 0.f && sim > THRED) ? fminf(sim, 1.f) : 0.f;
      sAdj[w * 256 + i * 16 + j] = val;
    }
  }

  // ---- Phase 5: neighbor-mean embedding output (independent of sAdj) ----
  for (int idx = tid; idx < 16 * 256; idx += 256) {
    int c = idx >> 8, d = idx & 255;
    float s = 0.f;
#pragma unroll
    for (int p = 0; p < 6; ++p) {
      s += sMask[p * 16 + c] * sB[p * 4160 + c * 260 + d];
    }
    out[((size_t)(b * CC + c)) * OUTW + d] = s * (1.f / 6.f);
  }
  __syncthreads();

  // ---- Phase 6: L1 row sums of adjacency (+ ones column) ----
  if (tid < 16) {
    float s = 1.f;
    for (int p = 0; p < 6; ++p) {
#pragma unroll
      for (int j = 0; j < 16; ++j) s += sAdj[p * 256 + tid * 16 + j];
    }
    sSum[tid] = 1.f / fmaxf(s, 1e-12f);
  }
  __syncthreads();

  // ---- Phase 7: normalized adjacency output (96 sims + ones column) ----
  for (int idx = tid; idx < 16 * 97; idx += 256) {
    int c = idx / 97, q = idx - c * 97;
    float inv = sSum[c];
    float v = (q == 96) ? inv
                        : sAdj[(q >> 4) * 256 + c * 16 + (q & 15)] * inv;
    out[((size_t)(b * CC + c)) * OUTW + DD + q] = v;
  }
}

extern "C" void kernel_launch(void* const* d_in, const int* in_sizes, int n_in,
                              void* d_out, int out_size, void* d_ws, size_t ws_size,
                              hipStream_t stream) {
  (void)in_sizes; (void)n_in; (void)out_size; (void)d_ws; (void)ws_size;
  const float* emb = (const float*)d_in[0];   // [B*C, D] f32
  const int*   nm  = (const int*)d_in[1];     // [B, C] int32 0/1
  float*       out = (float*)d_out;           // [B*C, 353] f32

  const size_t shmem = 30880 * sizeof(float); // 123520 bytes
  (void)hipFuncSetAttribute((const void*)graph_wmma_kernel,
                            hipFuncAttributeMaxDynamicSharedMemorySize,
                            (int)shmem);
  graph_wmma_kernel<<<dim3(BB), dim3(256), shmem, stream>>>(emb, nm, out);
}